// Conv2d_hebb_86122684220055
// MI455X (gfx1250) — compile-verified
//
#include <hip/hip_runtime.h>

#define B_     64
#define CIN    3
#define Hh     96
#define Ww     96
#define COUT   96
#define KH     5
#define KW     5
#define STRIDE 2
#define PAD    2
#define HOUT   48
#define WOUT   48
#define L_     (HOUT*WOUT)    /* 2304 */
#define K_     (CIN*KH*KW)    /* 75   */
#define KP     76             /* K padded to multiple of 4 */
#define LR     0.02f
#define DELTA  (-0.4f)
#define PRECF  1e-30f

typedef float v2f __attribute__((ext_vector_type(2)));
typedef float v8f __attribute__((ext_vector_type(8)));

// ---------------------------------------------------------------------------
// Kernel 1: fused dual-GEMM (out = W*pre, tot = (W*|W|)*pre) + top-2 + scatter
// Block: 192 threads (6 wave32), covers all 96 channels x 16 L-columns.
// ---------------------------------------------------------------------------
__global__ __launch_bounds__(192) void conv_hebb_gemm(
    const float* __restrict__ x, const float* __restrict__ w,
    float* __restrict__ out, float* __restrict__ yx, float* __restrict__ yy)
{
    __shared__ float lds_w[COUT * KP];   // weights, zero-padded K
    __shared__ float lds_p[KP * 16];     // im2col tile [k][n]
    __shared__ float lds_t[COUT * 16];   // tot tile   [o][n]
    __shared__ int   s_i0[16], s_i1[16];
    __shared__ float s_t0[16], s_t1[16];

    const int tid = threadIdx.x;
    const int b   = blockIdx.y;
    const int l0  = blockIdx.x * 16;

    // ---- stage weights into LDS (k=75 padded with zero) ----
    for (int i = tid; i < COUT * KP; i += 192) {
        int o = i / KP, k = i - o * KP;
        lds_w[i] = (k < K_) ? w[o * K_ + k] : 0.0f;
    }
    // ---- stage im2col tile [KP x 16] into LDS ----
    for (int i = tid; i < KP * 16; i += 192) {
        int k = i >> 4, n = i & 15;
        float v = 0.0f;
        if (k < K_) {
            int c  = k / (KH * KW);
            int r  = k - c * (KH * KW);
            int ki = r / KW, kj = r - ki * KW;
            int l  = l0 + n;
            int y  = l / WOUT, xo = l - y * WOUT;
            int iy = y * STRIDE + ki - PAD;
            int ix = xo * STRIDE + kj - PAD;
            if (iy >= 0 && iy < Hh && ix >= 0 && ix < Ww)
                v = x[((b * CIN + c) * Hh + iy) * Ww + ix];
        }
        lds_p[i] = v;
    }
    __syncthreads();

    const int wv   = tid >> 5;      // m-tile 0..5
    const int lane = tid & 31;
    const int mn   = lane & 15;     // A: row m / B,C,D: col n
    const int kg   = lane >> 4;     // k-group (0/1)

    v8f accO = {}; v8f accT = {};
    const float* wr = &lds_w[(wv * 16 + mn) * KP];
    #pragma unroll
    for (int kk = 0; kk < KP; kk += 4) {
        int ka = kk + 2 * kg;
        float a0 = wr[ka], a1 = wr[ka + 1];
        v2f aW = {a0, a1};
        v2f aL = {a0 * __builtin_fabsf(a0), a1 * __builtin_fabsf(a1)}; // sign(W)|W|^(p-1), p=3
        v2f bP = {lds_p[ka * 16 + mn], lds_p[(ka + 1) * 16 + mn]};
        accO = __builtin_amdgcn_wmma_f32_16x16x4_f32(
            false, aW, false, bP, (short)0, accO, false, false);
        accT = __builtin_amdgcn_wmma_f32_16x16x4_f32(
            false, aL, false, bP, (short)0, accT, false, false);
    }

    // C/D layout: VGPR r -> row (r + 8*kg), col = mn
    #pragma unroll
    for (int r = 0; r < 8; ++r) {
        int o = wv * 16 + r + 8 * kg;
        out[((size_t)b * COUT + o) * L_ + l0 + mn] = accO[r];
        lds_t[o * 16 + mn] = accT[r];
    }
    __syncthreads();

    // ---- top-2 over channels per column (earliest index wins ties) ----
    if (tid < 16) {
        int i0 = 0; float v0 = lds_t[tid];
        for (int o = 1; o < COUT; ++o) {
            float v = lds_t[o * 16 + tid];
            if (v > v0) { v0 = v; i0 = o; }
        }
        int i1 = (i0 == 0) ? 1 : 0;
        float v1 = lds_t[i1 * 16 + tid];
        for (int o = 0; o < COUT; ++o) {
            if (o == i0) continue;
            float v = lds_t[o * 16 + tid];
            if (v > v1) { v1 = v; i1 = o; }
        }
        s_i0[tid] = i0; s_i1[tid] = i1; s_t0[tid] = v0; s_t1[tid] = v1;
    }
    __syncthreads();

    // ---- scatter yx / yy (12 threads per column, 192 = 16*12) ----
    {
        int col = tid / 12;
        int r   = tid - col * 12;
        int i0 = s_i0[col], i1 = s_i1[col];
        for (int k = r; k < K_; k += 12) {
            float p = lds_p[k * 16 + col];
            atomicAdd(&yx[i0 * K_ + k], p);
            atomicAdd(&yx[i1 * K_ + k], DELTA * p);
        }
        if (r == 0) {
            atomicAdd(&yy[i0], s_t0[col]);
            atomicAdd(&yy[i1], DELTA * s_t1[col]);
        }
    }
}

// ---------------------------------------------------------------------------
// Workspace init (runs every launch -> deterministic)
// ---------------------------------------------------------------------------
__global__ void zero_ws(float* yx, float* yy, unsigned int* mb)
{
    int i = blockIdx.x * blockDim.x + threadIdx.x;
    if (i < COUT * K_) yx[i] = 0.0f;
    if (i < COUT)      yy[i] = 0.0f;
    if (i == 0)        *mb = __float_as_uint(PRECF);  // nc = max(max|ds|, PREC)
}

// ---------------------------------------------------------------------------
// ds = yx/L - (yy/L)*W ; nc = max(|ds|) via atomicMax on bits (ds >= 0 bits)
// ---------------------------------------------------------------------------
__global__ void finalize_ds(const float* __restrict__ w,
                            const float* __restrict__ yx,
                            const float* __restrict__ yy,
                            float* __restrict__ ds, unsigned int* mb)
{
    int i = blockIdx.x * blockDim.x + threadIdx.x;
    if (i < COUT * K_) {
        int o = i / K_;
        float d = yx[i] * (1.0f / (float)L_) - (yy[o] * (1.0f / (float)L_)) * w[i];
        ds[i] = d;
        atomicMax(mb, __float_as_uint(__builtin_fabsf(d)));
    }
}

__global__ void apply_update(const float* __restrict__ w,
                             const float* __restrict__ ds,
                             const unsigned int* __restrict__ mb,
                             float* __restrict__ neww)
{
    int i = blockIdx.x * blockDim.x + threadIdx.x;
    if (i < COUT * K_) {
        float nc = __uint_as_float(*mb);
        neww[i] = w[i] + LR * (ds[i] / nc);
    }
}

// ---------------------------------------------------------------------------
extern "C" void kernel_launch(void* const* d_in, const int* in_sizes, int n_in,
                              void* d_out, int out_size, void* d_ws, size_t ws_size,
                              hipStream_t stream)
{
    const float* x = (const float*)d_in[0];   // [64,3,96,96]
    const float* w = (const float*)d_in[1];   // [96,3,5,5]
    float* out  = (float*)d_out;                        // [64,96,48,48]
    float* neww = out + (size_t)B_ * COUT * L_;         // [96,75]

    float*        yx = (float*)d_ws;          // [96*75]
    float*        yy = yx + COUT * K_;        // [96]
    unsigned int* mb = (unsigned int*)(yy + COUT);
    float*        ds = (float*)(mb + 1);      // [96*75]

    zero_ws<<<(COUT * K_ + 255) / 256, 256, 0, stream>>>(yx, yy, mb);
    conv_hebb_gemm<<<dim3(L_ / 16, B_), 192, 0, stream>>>(x, w, out, yx, yy);
    finalize_ds<<<(COUT * K_ + 255) / 256, 256, 0, stream>>>(w, yx, yy, ds, mb);
    apply_update<<<(COUT * K_ + 255) / 256, 256, 0, stream>>>(w, ds, mb, neww);
}